// InverseConsistencyLoss_45715631899322
// MI455X (gfx1250) — compile-verified
//
#include <hip/hip_runtime.h>

// ---------------------------------------------------------------------------
// InverseConsistencyLoss for [B=2, C=3, 128,128,128] float32 fields.
//   coords = identity_grid + fwd
//   warped = trilinear_gather(bwd, coords)   (zero padding, pixel space)
//   loss   = nan_to_num(mean((fwd + warped)^2), nan=0, +inf=1000, -inf=0)
//
// Memory-bound gather kernel; WMMA (V_WMMA_F32_16X16X4_F32) is used as an
// exact-f32 deterministic wave32 cross-lane reduction primitive.
// ---------------------------------------------------------------------------

#define D_ 128
#define H_ 128
#define W_ 128
#define HW_ (H_ * W_)            // 16384
#define DHW_ (D_ * H_ * W_)      // 2097152 = 1<<21
#define B_ 2
#define NVOX (B_ * DHW_)         // 4194304 voxels, /32 and /262144 exact
#define NELEM (3LL * (long long)NVOX) // 12582912 elements in the mean

typedef __attribute__((ext_vector_type(2))) float v2f;
typedef __attribute__((ext_vector_type(8))) float v8f;

// Deterministic wave32 sum using two V_WMMA_F32_16X16X4_F32 ops.
// A layout (16x4 f32): lanes 0-15 -> A[m][0],A[m][1]; lanes 16-31 -> A[m][2],A[m][3].
// WMMA1: A = {x,0} per lane, B = ones(4x16)  => D[m][n] = x[m] + x[m+16]  (=r_m)
// C/D layout: vgpr j on lanes 0-15 holds r_j, on lanes 16-31 holds r_{j+8};
// summing the 8 result regs gives s_lo on lanes 0-15, s_hi on lanes 16-31.
// WMMA2: A = {s,0} => D[m][n] = s_lo + s_hi = total, broadcast to all lanes.
__device__ __forceinline__ float wave32_sum_wmma(float x) {
  v2f a; a.x = x;    a.y = 0.0f;
  v2f b; b.x = 1.0f; b.y = 1.0f;
  v8f c = {};
  v8f d = __builtin_amdgcn_wmma_f32_16x16x4_f32(
      /*neg_a=*/false, a, /*neg_b=*/false, b,
      /*c_mod=*/(short)0, c, /*reuse_a=*/false, /*reuse_b=*/false);
  float s = ((d[0] + d[1]) + (d[2] + d[3])) + ((d[4] + d[5]) + (d[6] + d[7]));
  v2f a2; a2.x = s; a2.y = 0.0f;
  v8f d2 = __builtin_amdgcn_wmma_f32_16x16x4_f32(
      false, a2, false, b, (short)0, c, false, false);
  return d2[0];
}

__device__ __forceinline__ int clamp127(int v) {
  return v < 0 ? 0 : (v > 127 ? 127 : v);
}

__global__ void __launch_bounds__(256)
icl_partial_kernel(const float* __restrict__ fwd,
                   const float* __restrict__ bwd,
                   float* __restrict__ part) {
  const int tid    = blockIdx.x * blockDim.x + threadIdx.x;
  const int stride = gridDim.x * blockDim.x;   // 262144; NVOX % stride == 0

  float acc = 0.0f;
  for (int v = tid; v < NVOX; v += stride) {   // exactly 16 uniform iters
    const int b = v >> 21;
    const int s = v & (DHW_ - 1);
    const int z = s >> 14;
    const int y = (s >> 7) & 127;
    const int x = s & 127;

    const float* f = fwd + (size_t)b * 3 * DHW_ + s;
    // Prefetch next stripe of the streamed forward field (global_prefetch_b8).
    if (v + stride < NVOX) {
      __builtin_prefetch(f + stride, 0, 1);
    }
    const float fz = f[0];
    const float fy = f[DHW_];
    const float fx = f[2 * DHW_];

    const float cz = (float)z + fz;
    const float cy = (float)y + fy;
    const float cx = (float)x + fx;

    const float z0f = floorf(cz), y0f = floorf(cy), x0f = floorf(cx);
    const float wz = cz - z0f, wy = cy - y0f, wx = cx - x0f;
    const int iz0 = (int)z0f, iy0 = (int)y0f, ix0 = (int)x0f;

    const float* bb = bwd + (size_t)b * 3 * DHW_;

    float a0 = 0.0f, a1 = 0.0f, a2 = 0.0f;
#pragma unroll
    for (int dz = 0; dz < 2; ++dz) {
      const int   zi  = iz0 + dz;
      const float wzc = dz ? wz : 1.0f - wz;
      const bool  vz  = (zi >= 0) & (zi < D_);
      const int   zc  = clamp127(zi);
#pragma unroll
      for (int dy = 0; dy < 2; ++dy) {
        const int   yi  = iy0 + dy;
        const float wyc = dy ? wy : 1.0f - wy;
        const bool  vy  = (yi >= 0) & (yi < H_);
        const int   yc  = clamp127(yi);
#pragma unroll
        for (int dx = 0; dx < 2; ++dx) {
          const int   xi  = ix0 + dx;
          const float wxc = dx ? wx : 1.0f - wx;
          const bool  vx  = (xi >= 0) & (xi < W_);
          const int   xc  = clamp127(xi);

          float w = wzc * wyc * wxc;
          w = (vz & vy & vx) ? w : 0.0f;

          const int idx = (zc * H_ + yc) * W_ + xc;
          a0 = fmaf(w, bb[idx],            a0);
          a1 = fmaf(w, bb[DHW_ + idx],     a1);
          a2 = fmaf(w, bb[2 * DHW_ + idx], a2);
        }
      }
    }

    const float c0 = fz + a0;
    const float c1 = fy + a1;
    const float c2 = fx + a2;
    acc += c0 * c0 + c1 * c1 + c2 * c2;
  }

  // Wave-level deterministic sum via WMMA (EXEC is all-ones here: the loop
  // trip count is uniform across the wave).
  const float wsum = wave32_sum_wmma(acc);

  __shared__ float lds[8];
  const int lane = threadIdx.x & 31;
  const int wave = threadIdx.x >> 5;
  if (lane == 0) lds[wave] = wsum;
  __syncthreads();
  if (threadIdx.x == 0) {
    float t = 0.0f;
#pragma unroll
    for (int i = 0; i < 8; ++i) t += lds[i];   // fixed order -> deterministic
    part[blockIdx.x] = t;
  }
}

// One wave (32 threads): fold 1024 block partials into the final scalar.
__global__ void __launch_bounds__(32)
icl_final_kernel(const float* __restrict__ part, float* __restrict__ out) {
  float s = 0.0f;
#pragma unroll 4
  for (int k = 0; k < 32; ++k) {
    s += part[threadIdx.x + 32 * k];           // fixed order per lane
  }
  const float tot = wave32_sum_wmma(s);
  if (threadIdx.x == 0) {
    float mean = tot * (1.0f / (float)NELEM);
    if (__builtin_isnan(mean)) {
      mean = 0.0f;
    } else if (__builtin_isinf(mean)) {
      mean = (mean > 0.0f) ? 1000.0f : 0.0f;
    }
    out[0] = mean;
  }
}

extern "C" void kernel_launch(void* const* d_in, const int* in_sizes, int n_in,
                              void* d_out, int out_size, void* d_ws, size_t ws_size,
                              hipStream_t stream) {
  const float* fwd = (const float*)d_in[0];   // forward_disp  [2,3,128,128,128]
  const float* bwd = (const float*)d_in[1];   // backward_disp [2,3,128,128,128]
  float* part = (float*)d_ws;                 // 1024 floats of scratch
  float* out  = (float*)d_out;                // 1 float

  icl_partial_kernel<<<1024, 256, 0, stream>>>(fwd, bwd, part);
  icl_final_kernel<<<1, 32, 0, stream>>>(part, out);
}